// AERC_31001073943129
// MI455X (gfx1250) — compile-verified
//
#include <hip/hip_runtime.h>

// CDNA5 / gfx1250, wave32. fp32 WMMA 16x16x4.
typedef __attribute__((ext_vector_type(2))) float v2f;
typedef __attribute__((ext_vector_type(8))) float v8f;

#define KDIM 512      // contraction length for all three GEMMs
#define KC   32       // K-chunk staged in LDS
#define LSTR 36       // KC + 4 pad: conflict-free strided ds_load_b64, 16B-aligned stores

__device__ __forceinline__ v8f wmma4(v2f a, v2f b, v8f c) {
  // 8-arg form: (neg_a, A, neg_b, B, c_mod, C, reuse_a, reuse_b)
  return __builtin_amdgcn_wmma_f32_16x16x4_f32(false, a, false, b, (short)0, c,
                                               false, false);
}

// ---------------------------------------------------------------------------
// C[t,c] = op(A(t,:) . Bm(c,:) + bias[c]),  A: MxK row-major, Bm: CxK row-major
// workgroup tile 128(t) x 256(c); 8 waves as 2x4; each wave 64x64 (4x4 WMMA tiles)
// grid = (cols/256, M/128)
// ---------------------------------------------------------------------------
template <bool RELU>
__global__ __launch_bounds__(256)
void gemm_bias(const float* __restrict__ A, const float* __restrict__ Bm,
               const float* __restrict__ bias, float* __restrict__ out, int ldc) {
  __shared__ float As[128 * LSTR];
  __shared__ float Bs[256 * LSTR];

  const int tid  = threadIdx.x;
  const int lane = tid & 31, w = tid >> 5;
  const int wt = w >> 2, wn = w & 3;        // 2 x 4 wave grid
  const int half = lane >> 4, lm = lane & 15;
  const int m0 = blockIdx.y * 128;
  const int c0 = blockIdx.x * 256;

  v8f C[4][4];
#pragma unroll
  for (int i = 0; i < 4; ++i)
#pragma unroll
    for (int j = 0; j < 4; ++j) C[i][j] = (v8f)0.0f;

  for (int kc = 0; kc < KDIM; kc += KC) {
#pragma unroll
    for (int i = 0; i < 4; ++i) {           // A chunk: 128x32 = 1024 float4
      int f = tid + 256 * i;
      int row = f >> 3, col = (f & 7) << 2;
      *(float4*)&As[row * LSTR + col] =
          *(const float4*)(A + (size_t)(m0 + row) * KDIM + kc + col);
    }
#pragma unroll
    for (int i = 0; i < 8; ++i) {           // B chunk: 256x32 = 2048 float4
      int f = tid + 256 * i;
      int row = f >> 3, col = (f & 7) << 2;
      *(float4*)&Bs[row * LSTR + col] =
          *(const float4*)(Bm + (size_t)(c0 + row) * KDIM + kc + col);
    }
    __syncthreads();

    const float* Ab = &As[(wt * 64 + lm) * LSTR + 2 * half];
    const float* Bb = &Bs[(wn * 64 + lm) * LSTR + 2 * half];
#pragma unroll
    for (int k4 = 0; k4 < KC; k4 += 4) {
      v2f a[4], b[4];
#pragma unroll
      for (int mt = 0; mt < 4; ++mt) a[mt] = *(const v2f*)(Ab + mt * 16 * LSTR + k4);
#pragma unroll
      for (int nt = 0; nt < 4; ++nt) b[nt] = *(const v2f*)(Bb + nt * 16 * LSTR + k4);
#pragma unroll
      for (int mt = 0; mt < 4; ++mt)
#pragma unroll
        for (int nt = 0; nt < 4; ++nt) C[mt][nt] = wmma4(a[mt], b[nt], C[mt][nt]);
    }
    __syncthreads();
  }

#pragma unroll
  for (int mt = 0; mt < 4; ++mt) {
    int tb = m0 + wt * 64 + mt * 16 + 8 * half;   // row of element j is tb + j
#pragma unroll
    for (int nt = 0; nt < 4; ++nt) {
      int c = c0 + wn * 64 + nt * 16 + lm;
      float bv = bias[c];
#pragma unroll
      for (int j = 0; j < 8; ++j) {
        float v = C[mt][nt][j] + bv;
        if (RELU) v = fmaxf(v, 0.0f);
        out[(size_t)(tb + j) * ldc + c] = v;
      }
    }
  }
}

// ---------------------------------------------------------------------------
// r_ol[t,h] = sum_n (X(t,:) . W2[h*512+n,:]) * R[t,n] + sum_n b2[h*512+n]*R[t,n]
// One workgroup per (t-block of 128, h). Full n=0..511 in 2 passes of 256,
// P tile stays in registers; reduced against R; no atomics (unique owner).
// grid = (T/128, H)
// ---------------------------------------------------------------------------
__global__ __launch_bounds__(256)
void attn_reduce(const float* __restrict__ X, const float* __restrict__ W2,
                 const float* __restrict__ b2, const float* __restrict__ R,
                 float* __restrict__ r_ol) {
  __shared__ float As[128 * LSTR];
  __shared__ float Bs[256 * LSTR];
  __shared__ float acc[128 * 4];

  const int tid  = threadIdx.x;
  const int lane = tid & 31, w = tid >> 5;
  const int wt = w >> 2, wn = w & 3;
  const int half = lane >> 4, lm = lane & 15;
  const int m0 = blockIdx.x * 128;
  const int h  = blockIdx.y;

  float part[4][8];
#pragma unroll
  for (int mt = 0; mt < 4; ++mt)
#pragma unroll
    for (int j = 0; j < 8; ++j) part[mt][j] = 0.0f;

  for (int pass = 0; pass < 2; ++pass) {
    const float* Bm = W2 + ((size_t)h * 512 + (size_t)pass * 256) * KDIM;
    v8f C[4][4];
#pragma unroll
    for (int i = 0; i < 4; ++i)
#pragma unroll
      for (int j = 0; j < 4; ++j) C[i][j] = (v8f)0.0f;

    for (int kc = 0; kc < KDIM; kc += KC) {
#pragma unroll
      for (int i = 0; i < 4; ++i) {
        int f = tid + 256 * i;
        int row = f >> 3, col = (f & 7) << 2;
        *(float4*)&As[row * LSTR + col] =
            *(const float4*)(X + (size_t)(m0 + row) * KDIM + kc + col);
      }
#pragma unroll
      for (int i = 0; i < 8; ++i) {
        int f = tid + 256 * i;
        int row = f >> 3, col = (f & 7) << 2;
        *(float4*)&Bs[row * LSTR + col] =
            *(const float4*)(Bm + (size_t)row * KDIM + kc + col);
      }
      __syncthreads();

      const float* Ab = &As[(wt * 64 + lm) * LSTR + 2 * half];
      const float* Bb = &Bs[(wn * 64 + lm) * LSTR + 2 * half];
#pragma unroll
      for (int k4 = 0; k4 < KC; k4 += 4) {
        v2f a[4], b[4];
#pragma unroll
        for (int mt = 0; mt < 4; ++mt) a[mt] = *(const v2f*)(Ab + mt * 16 * LSTR + k4);
#pragma unroll
        for (int nt = 0; nt < 4; ++nt) b[nt] = *(const v2f*)(Bb + nt * 16 * LSTR + k4);
#pragma unroll
        for (int mt = 0; mt < 4; ++mt)
#pragma unroll
          for (int nt = 0; nt < 4; ++nt) C[mt][nt] = wmma4(a[mt], b[nt], C[mt][nt]);
      }
      __syncthreads();
    }

    // fold P * R into per-lane partials (lane lm owns column n = ... + lm)
#pragma unroll
    for (int mt = 0; mt < 4; ++mt) {
      int tb = m0 + wt * 64 + mt * 16 + 8 * half;
#pragma unroll
      for (int nt = 0; nt < 4; ++nt) {
        int n = pass * 256 + wn * 64 + nt * 16 + lm;
#pragma unroll
        for (int j = 0; j < 8; ++j)
          part[mt][j] += C[mt][nt][j] * R[(size_t)(tb + j) * KDIM + n];
      }
    }
  }

  // reduce 16-lane halves (each half holds one t, 16 consecutive n's)
#pragma unroll
  for (int mt = 0; mt < 4; ++mt)
#pragma unroll
    for (int j = 0; j < 8; ++j) {
      float v = part[mt][j];
      v += __shfl_xor(v, 1, 32);
      v += __shfl_xor(v, 2, 32);
      v += __shfl_xor(v, 4, 32);
      v += __shfl_xor(v, 8, 32);
      if (lm == 0) acc[(wt * 64 + mt * 16 + j + 8 * half) * 4 + wn] = v;
    }
  __syncthreads();

  if (tid < 128) {
    int t = m0 + tid;
    float s = acc[tid * 4 + 0] + acc[tid * 4 + 1] + acc[tid * 4 + 2] + acc[tid * 4 + 3];
    const float* b2h = b2 + (size_t)h * 512;
    const float* rt  = R + (size_t)t * KDIM;
    float s0 = 0.f, s1 = 0.f, s2 = 0.f, s3 = 0.f;
    for (int n = 0; n < 512; n += 4) {
      float4 bb = *(const float4*)(b2h + n);
      float4 rr = *(const float4*)(rt + n);
      s0 += bb.x * rr.x; s1 += bb.y * rr.y; s2 += bb.z * rr.z; s3 += bb.w * rr.w;
    }
    r_ol[(size_t)t * 512 + h] = s + (s0 + s1) + (s2 + s3);
  }
}

// ---------------------------------------------------------------------------
extern "C" void kernel_launch(void* const* d_in, const int* in_sizes, int n_in,
                              void* d_out, int out_size, void* d_ws, size_t ws_size,
                              hipStream_t stream) {
  const float* r    = (const float*)d_in[0];  // (1024, 512)
  const float* W1   = (const float*)d_in[1];  // (512, 512)
  const float* b1   = (const float*)d_in[2];  // (512)
  const float* W2   = (const float*)d_in[3];  // (262144, 512)
  const float* b2   = (const float*)d_in[4];  // (262144)
  const float* Wout = (const float*)d_in[5];  // (32000, 512)
  const float* bout = (const float*)d_in[6];  // (32000)
  float* logits = (float*)d_out;              // (1024, 32000)

  float* x    = (float*)d_ws;                 // 1024*512 f32 = 2 MB
  float* r_ol = (float*)d_ws + 1024 * 512;    // 1024*512 f32 = 2 MB

  // x = relu(r @ W1^T + b1)        grid: cols 512/256 x rows 1024/128
  gemm_bias<true><<<dim3(2, 8), 256, 0, stream>>>(r, W1, b1, x, 512);

  // r_ol[t,h] = r_t^T W2_h x_t + b2_h . r_t   (fused, W_att never materialized)
  attn_reduce<<<dim3(8, 512), 256, 0, stream>>>(x, W2, b2, r, r_ol);

  // logits = r_ol @ W_out^T + b_out   grid: 32000/256 x 1024/128
  gemm_bias<false><<<dim3(125, 8), 256, 0, stream>>>(r_ol, Wout, bout, logits, 32000);
}